// REM_AttentionModule_80504866996662
// MI455X (gfx1250) — compile-verified
//
#include <hip/hip_runtime.h>
#include <hip/hip_bf16.h>

// Problem constants from the reference: B=8, S=2048, D=1024, f32 throughout.
namespace {
constexpr int kB = 8;
constexpr int kS = 2048;
constexpr int kD = 1024;
constexpr int kSChunks = 32;              // S-split for the streaming reduction
constexpr int kSPerChunk = kS / kSChunks; // 64 rows per block
constexpr int kMPad = 16;                 // pad M=8 -> 16 for the 16x16 WMMA tile
}

typedef float v2f __attribute__((ext_vector_type(2)));
typedef float v8f __attribute__((ext_vector_type(8)));

// ---------------------------------------------------------------------------
// Kernel 1: partial column sum of x over S.  x is [B,S,D]; each block owns
// (batch b, S-chunk c) and streams 64 rows of 1024 floats with float4 loads
// (perfectly coalesced: 256 threads cover one full 4 KiB row per iteration).
// Writes partial[b][c][0..D) -- no atomics, fully deterministic.
// ---------------------------------------------------------------------------
__global__ __launch_bounds__(256) void rem_attn_colsum_partial(
    const float* __restrict__ x, float* __restrict__ part) {
  const int blk = blockIdx.x;            // 0 .. B*kSChunks-1
  const int b = blk / kSChunks;
  const int c = blk % kSChunks;
  const int t = threadIdx.x;             // float4 column index, 0..255

  const float4* __restrict__ xr =
      (const float4*)(x + (size_t)b * kS * kD + (size_t)c * kSPerChunk * kD);

  float4 acc = make_float4(0.f, 0.f, 0.f, 0.f);
#pragma unroll 4
  for (int s = 0; s < kSPerChunk; ++s) {
    float4 v = xr[(size_t)s * (kD / 4) + t];
    acc.x += v.x; acc.y += v.y; acc.z += v.z; acc.w += v.w;
  }
  ((float4*)(part + ((size_t)b * kSChunks + c) * kD))[t] = acc;
}

// ---------------------------------------------------------------------------
// Kernel 2: fold the kSChunks partials into Xsum[16][1024].
// Rows 8..15 are written as zero so the WMMA A-matrix padding rows are clean.
// ---------------------------------------------------------------------------
__global__ __launch_bounds__(256) void rem_attn_colsum_final(
    const float* __restrict__ part, float* __restrict__ xsum) {
  const int idx = blockIdx.x * blockDim.x + threadIdx.x; // 0 .. 16*1024-1
  const int m = idx / kD;
  const int d = idx % kD;
  float acc = 0.f;
  if (m < kB) {
#pragma unroll
    for (int c = 0; c < kSChunks; ++c)
      acc += part[((size_t)m * kSChunks + c) * kD + d];
  }
  xsum[idx] = acc;
}

// ---------------------------------------------------------------------------
// Kernel 3: out[m][n] = sum_k Xsum[m][k] * Wv[n][k] + 2048*bv[n]
// via V_WMMA_F32_16X16X4_F32.  One 16x16 output tile per wave32; 64 waves.
//
// ISA VGPR layouts (CDNA5 ISA):
//   A 16x4 f32 : lane L -> M = L%16;  VGPR0 = K=k+2*(L/16), VGPR1 = that+1
//   B 4x16 f32 : lane L -> N = L%16;  same K split across lane halves
//   C/D 16x16  : VGPR r, lanes 0-15 -> M=r, lanes 16-31 -> M=r+8
// Both A and B fragments are contiguous float2 loads per lane.
// ---------------------------------------------------------------------------
__global__ __launch_bounds__(256) void rem_attn_wmma_gemm(
    const float* __restrict__ xsum, const float* __restrict__ Wv,
    const float* __restrict__ bv, float* __restrict__ out) {
  const int wave = (blockIdx.x * blockDim.x + threadIdx.x) >> 5; // 0..63
  const int lane = threadIdx.x & 31;
  const int n0   = wave * 16;          // output-feature tile base
  const int l16  = lane & 15;
  const int half = lane >> 4;          // 0: K+{0,1}, 1: K+{2,3}
  const int kb   = half * 2;

  const float* __restrict__ arow = xsum + (size_t)l16 * kD;        // M = l16
  const float* __restrict__ brow = Wv + (size_t)(n0 + l16) * kD;   // N = l16

  v8f c0 = {0.f, 0.f, 0.f, 0.f, 0.f, 0.f, 0.f, 0.f};
  v8f c1 = {0.f, 0.f, 0.f, 0.f, 0.f, 0.f, 0.f, 0.f};

#pragma unroll 4
  for (int k = 0; k < kD; k += 8) {
    v2f a0 = *(const v2f*)(arow + k + kb);
    v2f b0 = *(const v2f*)(brow + k + kb);
    v2f a1 = *(const v2f*)(arow + k + 4 + kb);
    v2f b1 = *(const v2f*)(brow + k + 4 + kb);
    // 8 args: (neg_a, A, neg_b, B, c_mod, C, reuse_a, reuse_b)
    c0 = __builtin_amdgcn_wmma_f32_16x16x4_f32(false, a0, false, b0,
                                               (short)0, c0, false, false);
    c1 = __builtin_amdgcn_wmma_f32_16x16x4_f32(false, a1, false, b1,
                                               (short)0, c1, false, false);
  }
  v8f c = c0 + c1;

  const float bias = 2048.0f * bv[n0 + l16];  // S * bv (attention rows == 1)

  // Lanes 0-15 of VGPR r hold M=r (batch rows 0..7); lanes 16-31 hold the
  // zero-padded rows 8..15 and are discarded.  Divergence only after WMMA.
  if (half == 0) {
#pragma unroll
    for (int r = 0; r < 8; ++r) {
      out[(size_t)r * kD + n0 + l16] = c[r] + bias;
    }
  }
}

// ---------------------------------------------------------------------------
// Host launcher.  Inputs (setup_inputs order):
//   0: tensor_sequence [8,2048,1024] f32   5: Wv [1024,1024] f32
//   6: bv [1024] f32          (Wq/bq/Wk/bk are mathematically dead)
// d_out: [8,1024] f32.
// Workspace: Xsum 16*1024 f32 (64 KiB) then partials 8*32*1024 f32 (1 MiB).
// ---------------------------------------------------------------------------
extern "C" void kernel_launch(void* const* d_in, const int* in_sizes, int n_in,
                              void* d_out, int out_size, void* d_ws, size_t ws_size,
                              hipStream_t stream) {
  (void)in_sizes; (void)n_in; (void)out_size; (void)ws_size;

  const float* x  = (const float*)d_in[0];
  const float* Wv = (const float*)d_in[5];
  const float* bv = (const float*)d_in[6];
  float* out = (float*)d_out;

  float* xsum = (float*)d_ws;                 // [16][1024]
  float* part = xsum + (size_t)kMPad * kD;    // [8][32][1024]

  rem_attn_colsum_partial<<<kB * kSChunks, 256, 0, stream>>>(x, part);
  rem_attn_colsum_final<<<(kMPad * kD) / 256, 256, 0, stream>>>(part, xsum);
  rem_attn_wmma_gemm<<<(64 * 32) / 256, 256, 0, stream>>>(xsum, Wv, bv, out);
}